// ModelBlockDiag_79336635892142
// MI455X (gfx1250) — compile-verified
//
#include <hip/hip_runtime.h>

// CDNA5 / gfx1250: wave32, WMMA f32 16x16x4 (D = A(16x4) * B(4x16) + C)
typedef __attribute__((ext_vector_type(2))) float v2f;
typedef __attribute__((ext_vector_type(4))) float v4f;
typedef __attribute__((ext_vector_type(8))) float v8f;
typedef __attribute__((ext_vector_type(4))) int   v4i;

#define KC 32           // K chunk staged in LDS (8 wmma steps)
#define LDSROW 36       // padded dword stride per n-row (bank-conflict-free b128 reads)
#define NT 4            // 4 n-tiles of 16 -> 64 columns per wave
#define MT 2            // 2 m-tiles of 16 -> 32 rows per wave
#define WAVES_PER_WG 8  // 256 threads
#define M_PER_WG (WAVES_PER_WG * MT * 16)  // 256 rows per workgroup

#define AS1 __attribute__((address_space(1)))
#define AS3 __attribute__((address_space(3)))

// Async global -> LDS copy of 16 bytes per lane (gfx1250, tracked by ASYNCcnt).
__device__ __forceinline__ void async_load_b128(const float* gptr, float* lptr)
{
#if __has_builtin(__builtin_amdgcn_global_load_async_to_lds_b128)
    __builtin_amdgcn_global_load_async_to_lds_b128(
        (AS1 v4i*)gptr, (AS3 v4i*)lptr, 0, 0);
#else
    unsigned loff = (unsigned)(size_t)(AS3 void*)lptr;   // LDS byte offset
    asm volatile("global_load_async_to_lds_b128 %0, %1, off"
                 :: "v"(loff), "v"(gptr) : "memory");
#endif
}

__device__ __forceinline__ void wait_async0()
{
#if __has_builtin(__builtin_amdgcn_s_wait_asynccnt)
    __builtin_amdgcn_s_wait_asynccnt(0);
#else
    asm volatile("s_wait_asynccnt 0x0" ::: "memory");
#endif
}

// Block-diagonal layer: for parallel block p = blockIdx.z,
// C[m, cOffPerP*p + n] = act( sum_k A[m, aOffPerP*p+k] * W[wRowPerP*p+n, wColPerP*p+k] + bias[biasPerP*p+n] )
//
// K-slot mapping (identical for A and B, so the product is exact):
// within an 8-K group at base kb, wmma step s in {0,1} uses
//   k = kb + 4*half + 2*s + {0,1}   (half = lane>>4)
// so one b128 load per lane covers both steps of a group.
__global__ __launch_bounds__(256)
void mlp_layer_wmma(const float* __restrict__ A, int ldA, int aOffPerP,
                    const float* __restrict__ W, int ldW, int wRowPerP, int wColPerP,
                    const float* __restrict__ bias, int biasPerP,
                    float* __restrict__ C, int ldC, int cOffPerP,
                    int K, int doRelu)
{
    __shared__ float ldsB[2][64 * LDSROW];   // double-buffered 64(N) x 32(K) tile, 18 KB

    const int t    = threadIdx.x;
    const int lane = t & 31;
    const int wave = t >> 5;
    const int l    = lane & 15;   // A row within tile / B col within tile
    const int half = lane >> 4;   // K-group selector

    const int p  = blockIdx.z;
    const int m0 = blockIdx.x * M_PER_WG + wave * (MT * 16);
    const int n0 = blockIdx.y * (16 * NT);

    // ---- cooperative B staging mapping: 64 rows x 32 dwords, 8 dwords/thread
    const int sr = t >> 2;          // 0..63 : n-row within tile
    const int sc = (t & 3) * 8;     // dword offset: 0,8,16,24
    const float* wrow = W + (size_t)(wRowPerP * p + n0 + sr) * ldW
                          + (size_t)wColPerP * p;

    // ---- per-lane A fragment base pointers (one per m-tile)
    const float* aBase[MT];
#pragma unroll
    for (int mf = 0; mf < MT; ++mf)
        aBase[mf] = A + (size_t)(m0 + mf * 16 + l) * ldA
                      + (size_t)aOffPerP * p + 4 * half;

    v8f acc[MT][NT];
#pragma unroll
    for (int mf = 0; mf < MT; ++mf)
#pragma unroll
        for (int j = 0; j < NT; ++j) acc[mf][j] = (v8f)0.0f;

    const int nc = K / KC;

    // ---- prologue: async-stage chunk 0 into buffer 0
    async_load_b128(wrow + sc,     &ldsB[0][sr * LDSROW + sc]);
    async_load_b128(wrow + sc + 4, &ldsB[0][sr * LDSROW + sc + 4]);
    wait_async0();
    __syncthreads();

    for (int c = 0; c < nc; ++c) {
        const int cK  = c * KC;
        const int buf = c & 1;
        const bool hasNext = (c + 1) < nc;

        // issue async loads for next chunk's B tile; they complete in the
        // background (ASYNCcnt) while this chunk's 64 wmmas run
        if (hasNext) {
            const int nbuf = (c + 1) & 1;
            async_load_b128(wrow + cK + KC + sc,
                            &ldsB[nbuf][sr * LDSROW + sc]);
            async_load_b128(wrow + cK + KC + sc + 4,
                            &ldsB[nbuf][sr * LDSROW + sc + 4]);
        }

        // prefetch-hint the chunk after next (L2)
        if (c + 2 < nc)
            __builtin_prefetch(wrow + cK + 2 * KC + sc, 0, 1);

        // load all A fragments for this chunk (4 groups x MT), b128 each
        v4f afr[4][MT];
#pragma unroll
        for (int g = 0; g < 4; ++g)
#pragma unroll
            for (int mf = 0; mf < MT; ++mf)
                afr[g][mf] = *(const v4f*)(aBase[mf] + cK + g * 8);

        // ---- compute: 4 groups x 2 steps x (MT x NT) wmmas = 64 per chunk
#pragma unroll
        for (int g = 0; g < 4; ++g) {
            v4f bfr[NT];
#pragma unroll
            for (int j = 0; j < NT; ++j)
                bfr[j] = *(const v4f*)&ldsB[buf][(16 * j + l) * LDSROW + g * 8 + 4 * half];
#pragma unroll
            for (int s = 0; s < 2; ++s) {
#pragma unroll
                for (int mf = 0; mf < MT; ++mf) {
                    v2f a2;
                    a2.x = afr[g][mf][2 * s];
                    a2.y = afr[g][mf][2 * s + 1];
#pragma unroll
                    for (int j = 0; j < NT; ++j) {
                        v2f b2;
                        b2.x = bfr[j][2 * s];
                        b2.y = bfr[j][2 * s + 1];
                        acc[mf][j] = __builtin_amdgcn_wmma_f32_16x16x4_f32(
                            false, a2, false, b2, (short)0, acc[mf][j], false, false);
                    }
                }
            }
        }

        // my async loads for the next chunk are done; after the barrier,
        // everyone's are done AND everyone has finished reading ldsB[buf],
        // so iteration c+1 may both read nbuf and async-overwrite buf.
        wait_async0();
        __syncthreads();
    }

    // ---- writeback: D VGPR i holds row m0 + mf*16 + 8*half + i, col n0+16j+l
#pragma unroll
    for (int mf = 0; mf < MT; ++mf) {
#pragma unroll
        for (int j = 0; j < NT; ++j) {
            int n    = n0 + 16 * j + l;
            int gcol = cOffPerP * p + n;
            float bv = bias[biasPerP * p + n];
#pragma unroll
            for (int i = 0; i < 8; ++i) {
                int m   = m0 + mf * 16 + 8 * half + i;
                float v = acc[mf][j][i] + bv;
                if (doRelu) v = v > 0.0f ? v : 0.0f;
                C[(size_t)m * ldC + gcol] = v;
            }
        }
    }
}

// x_p[b, p*256 + i] = x[b, i]  for p in 0..7
__global__ __launch_bounds__(256)
void tile_x_kernel(const float* __restrict__ x, float* __restrict__ xp)
{
    int b = blockIdx.x;
    int i = threadIdx.x;            // 0..255
    float v = x[(size_t)b * 256 + i];
    float* o = xp + (size_t)b * 2048;
#pragma unroll
    for (int p = 0; p < 8; ++p) o[p * 256 + i] = v;
}

extern "C" void kernel_launch(void* const* d_in, const int* in_sizes, int n_in,
                              void* d_out, int out_size, void* d_ws, size_t ws_size,
                              hipStream_t stream)
{
    const float* x  = (const float*)d_in[0];   // [1024, 256]
    const float* W0 = (const float*)d_in[1];   // [4096, 2048]
    const float* b0 = (const float*)d_in[2];   // [4096]
    const float* W1 = (const float*)d_in[3];   // [4096, 4096]
    const float* b1 = (const float*)d_in[4];
    const float* W2 = (const float*)d_in[5];   // [4096, 4096]
    const float* b2 = (const float*)d_in[6];
    const float* W3 = (const float*)d_in[7];   // [2048, 4096]
    const float* b3 = (const float*)d_in[8];   // [2048]

    float* y  = (float*)d_out;                 // [1024, 2048]
    float* xp = y + (size_t)1024 * 2048;       // [1024, 2048]

    float* hA = (float*)d_ws;                  // [1024, 4096]
    float* hB = hA + (size_t)1024 * 4096;      // [1024, 4096]

    dim3 blk(256, 1, 1);

    // x_p output (tuple element 2)
    tile_x_kernel<<<dim3(1024, 1, 1), blk, 0, stream>>>(x, xp);

    // L0: h = relu(x @ W0_p.T + b0), per-block K=256, N=512
    mlp_layer_wmma<<<dim3(4, 8, 8), blk, 0, stream>>>(
        x, 256, 0,
        W0, 2048, 512, 256,
        b0, 512,
        hA, 4096, 512,
        256, 1);

    // L1: h = relu(h @ W1_p.T + b1), K=512, N=512
    mlp_layer_wmma<<<dim3(4, 8, 8), blk, 0, stream>>>(
        hA, 4096, 512,
        W1, 4096, 512, 512,
        b1, 512,
        hB, 4096, 512,
        512, 1);

    // L2
    mlp_layer_wmma<<<dim3(4, 8, 8), blk, 0, stream>>>(
        hB, 4096, 512,
        W2, 4096, 512, 512,
        b2, 512,
        hA, 4096, 512,
        512, 1);

    // L3: y = h @ W3_p.T + b3 (no relu), K=512, N=256
    mlp_layer_wmma<<<dim3(4, 4, 8), blk, 0, stream>>>(
        hA, 4096, 512,
        W3, 4096, 256, 512,
        b3, 256,
        y, 2048, 256,
        512, 0);
}